// FCOSLoss_73400991089041
// MI455X (gfx1250) — compile-verified
//
#include <hip/hip_runtime.h>
#include <hip/hip_bf16.h>

typedef __attribute__((ext_vector_type(2))) float v2f;
typedef __attribute__((ext_vector_type(8))) float v8f;

#define INF_F   100000000.0f
#define TOTAL   172032      // 8 * (128*128 + 64*64 + 32*32)
#define PER_IMG 21504
#define THREADS 256
#define NBLK    704         // ceil(172032/256)=672, padded to multiple of 64 for WMMA

__device__ __forceinline__ float softplus_f(float x) {
    // log(1 + exp(x)) == logaddexp(0, x), numerically stable
    return fmaxf(x, 0.f) + log1pf(expf(-fabsf(x)));
}

__global__ __launch_bounds__(THREADS) void fcos_point_kernel(
    const float* __restrict__ cls0, const float* __restrict__ cls1, const float* __restrict__ cls2,
    const float* __restrict__ reg0, const float* __restrict__ reg1, const float* __restrict__ reg2,
    const float* __restrict__ ctr0, const float* __restrict__ ctr1, const float* __restrict__ ctr2,
    const float* __restrict__ boxes, const int* __restrict__ labels,
    float* __restrict__ partials)
{
    const int r = blockIdx.x * THREADS + threadIdx.x;
    float c_cls = 0.f, c_giou = 0.f, c_bce = 0.f, c_pos = 0.f;

    if (r < TOTAL) {
        // ---------- target side: image-major flatten (b, then level, then pixel) ----------
        const int bt = r / PER_IMG;
        const int q  = r - bt * PER_IMG;
        float px, py, lmin, lmax;
        if (q < 16384)      { int p = q;         py = (float)(p >> 7) * 4.f  + 2.f; px = (float)(p & 127) * 4.f  + 2.f; lmin = -1.f;  lmax = 64.f;  }
        else if (q < 20480) { int p = q - 16384; py = (float)(p >> 6) * 8.f  + 4.f; px = (float)(p & 63)  * 8.f  + 4.f; lmin = 64.f;  lmax = 128.f; }
        else                { int p = q - 20480; py = (float)(p >> 5) * 16.f + 8.f; px = (float)(p & 31)  * 16.f + 8.f; lmin = 128.f; lmax = INF_F; }

        float best_area = INF_F;
        int   best_idx  = 0;
        float btl = 0.f, btt = 0.f, btr = 0.f, btb = 0.f;
        const float* bx = boxes + (size_t)bt * 32 * 4;
        #pragma unroll 4
        for (int m = 0; m < 32; ++m) {
            const float b0 = bx[m*4+0], b1 = bx[m*4+1], b2 = bx[m*4+2], b3 = bx[m*4+3];
            const float l  = px - b0, t = py - b1, rr = b2 - px, bb = b3 - py;
            const float mn = fminf(fminf(l, t), fminf(rr, bb));
            const float mx = fmaxf(fmaxf(l, t), fmaxf(rr, bb));
            const bool  ok = (mn > 0.f) && (mx >= lmin) && (mx <= lmax);
            const float area = (b2 - b0) * (b3 - b1);
            const float am   = ok ? area : INF_F;
            if (m == 0 || am < best_area) {   // jnp.argmin: first occurrence of min
                best_area = am; best_idx = m;
                btl = l; btt = t; btr = rr; btb = bb;
            }
        }
        const bool pos = best_area < INF_F;
        const int  tc  = pos ? labels[bt * 32 + best_idx] : -1;
        float ctr_t = 0.f;
        if (pos) {
            const float lrmin = fminf(btl, btr), lrmax = fmaxf(btl, btr);
            const float tbmin = fminf(btt, btb), tbmax = fmaxf(btt, btb);
            const float ratio = (lrmin / fmaxf(lrmax, 1e-12f)) * (tbmin / fmaxf(tbmax, 1e-12f));
            ctr_t = sqrtf(fmaxf(ratio, 0.f));
        }

        // ---------- prediction side: level-major flatten (level, then b, then pixel) ----------
        const float* clsP; const float* regP; const float* ctrP; int hw, bp, pix;
        if (r < 131072)      {               clsP = cls0; regP = reg0; ctrP = ctr0; hw = 16384; bp = r >> 14; pix = r & 16383; }
        else if (r < 163840) { int u = r - 131072; clsP = cls1; regP = reg1; ctrP = ctr1; hw = 4096; bp = u >> 12; pix = u & 4095; }
        else                 { int u = r - 163840; clsP = cls2; regP = reg2; ctrP = ctr2; hw = 1024; bp = u >> 10; pix = u & 1023; }

        // sigmoid focal loss over 20 classes
        float maxp = 0.f, fsum = 0.f;
        const float* cbase = clsP + (size_t)bp * 20 * hw + pix;
        #pragma unroll
        for (int c = 0; c < 20; ++c) {
            const float x   = cbase[c * hw];
            const float p   = 1.f / (1.f + expf(-x));
            maxp = fmaxf(maxp, p);
            const float oh  = (pos && c == tc) ? 1.f : 0.f;
            const float ce  = softplus_f(x) - x * oh;
            const float p_t = p * oh + (1.f - p) * (1.f - oh);
            const float a_t = 0.25f * oh + 0.75f * (1.f - oh);
            const float om  = 1.f - p_t;
            fsum += a_t * ce * om * om;
        }
        const float w = ((tc == -1) && (maxp > 0.3f)) ? 0.f : 1.f;
        c_cls = fsum * w;

        // GIoU * centerness + centerness BCE (both vanish for negatives via posf mask)
        if (pos) {
            const float* rbase = regP + (size_t)bp * 4 * hw + pix;
            const float pl = rbase[0], pt = rbase[hw], pr = rbase[2*hw], pb = rbase[3*hw];
            const float t_area = (btl + btr) * (btt + btb);
            const float p_area = (pl + pr) * (pt + pb);
            const float w_i = fminf(pl, btl) + fminf(pr, btr);
            const float h_i = fminf(pb, btb) + fminf(pt, btt);
            const float a_i = w_i * h_i;
            const float a_u = t_area + p_area - a_i;
            const float ious = (a_i + 1.f) / (a_u + 1.f);
            const float gw = fmaxf(pl, btl) + fmaxf(pr, btr);
            const float gh = fmaxf(pb, btb) + fmaxf(pt, btt);
            const float ac = gw * gh;
            const float gious = ious - (ac - a_u) / ac;
            c_giou = (1.f - gious) * ctr_t;

            const float fc = ctrP[(size_t)bp * hw + pix];
            c_bce = softplus_f(fc) - fc * ctr_t;
            c_pos = 1.f;
        }
    }

    // ---------- deterministic per-block reduction ----------
    __shared__ float sm[4][THREADS];
    sm[0][threadIdx.x] = c_cls;
    sm[1][threadIdx.x] = c_giou;
    sm[2][threadIdx.x] = c_bce;
    sm[3][threadIdx.x] = c_pos;
    __syncthreads();
    for (int s = THREADS / 2; s > 0; s >>= 1) {
        if (threadIdx.x < (unsigned)s) {
            sm[0][threadIdx.x] += sm[0][threadIdx.x + s];
            sm[1][threadIdx.x] += sm[1][threadIdx.x + s];
            sm[2][threadIdx.x] += sm[2][threadIdx.x + s];
            sm[3][threadIdx.x] += sm[3][threadIdx.x + s];
        }
        __syncthreads();
    }
    if (threadIdx.x == 0) {
        partials[0 * NBLK + blockIdx.x] = sm[0][0];
        partials[1 * NBLK + blockIdx.x] = sm[1][0];
        partials[2 * NBLK + blockIdx.x] = sm[2][0];
        partials[3 * NBLK + blockIdx.x] = sm[3][0];
    }
}

// Single-wave finalize: reduce 704 partials per quantity with V_WMMA_F32_16X16X4_F32.
// D = A x Ones(4x16) + C accumulates row-sums; every one of the 16 columns of the
// accumulator holds an identical copy, so total = (sum of all D elements) / 16.
__global__ __launch_bounds__(32) void fcos_finalize_kernel(
    const float* __restrict__ partials, float* __restrict__ out)
{
    const int lane = threadIdx.x;   // exactly one wave32, EXEC all ones
    float totals[4];
    #pragma unroll
    for (int qd = 0; qd < 4; ++qd) {
        v8f acc = {};
        const float* base = partials + qd * NBLK;
        #pragma unroll
        for (int ch = 0; ch < NBLK / 64; ++ch) {
            v2f a;
            a.x = base[ch * 64 + 2 * lane + 0];
            a.y = base[ch * 64 + 2 * lane + 1];
            v2f b; b.x = 1.f; b.y = 1.f;   // B = ones(4x16)
            acc = __builtin_amdgcn_wmma_f32_16x16x4_f32(
                false, a, false, b, (short)0, acc, false, false);
        }
        float s = acc[0] + acc[1] + acc[2] + acc[3] + acc[4] + acc[5] + acc[6] + acc[7];
        #pragma unroll
        for (int off = 16; off > 0; off >>= 1)
            s += __shfl_xor(s, off, 32);
        totals[qd] = s * (1.f / 16.f);
    }
    if (lane == 0) {
        const float num_pos = fmaxf(totals[3], 1.f);
        out[0] = totals[0] / num_pos;   // loss_cls
        out[1] = totals[1] / num_pos;   // loss_bbox
        out[2] = totals[2] / num_pos;   // loss_centerness
    }
}

extern "C" void kernel_launch(void* const* d_in, const int* in_sizes, int n_in,
                              void* d_out, int out_size, void* d_ws, size_t ws_size,
                              hipStream_t stream) {
    (void)in_sizes; (void)n_in; (void)out_size; (void)ws_size;
    const float* cls0 = (const float*)d_in[0];
    const float* cls1 = (const float*)d_in[1];
    const float* cls2 = (const float*)d_in[2];
    const float* reg0 = (const float*)d_in[3];
    const float* reg1 = (const float*)d_in[4];
    const float* reg2 = (const float*)d_in[5];
    const float* ctr0 = (const float*)d_in[6];
    const float* ctr1 = (const float*)d_in[7];
    const float* ctr2 = (const float*)d_in[8];
    const float* boxes  = (const float*)d_in[9];
    const int*   labels = (const int*)d_in[10];

    float* partials = (float*)d_ws;   // 4 * NBLK floats = 11264 bytes

    fcos_point_kernel<<<NBLK, THREADS, 0, stream>>>(
        cls0, cls1, cls2, reg0, reg1, reg2, ctr0, ctr1, ctr2,
        boxes, labels, partials);
    fcos_finalize_kernel<<<1, 32, 0, stream>>>(partials, (float*)d_out);
}